// Tabulated_4647154614863
// MI455X (gfx1250) — compile-verified
//
#include <hip/hip_runtime.h>
#include <hip/hip_bf16.h>

#define N_TABLE 1024
#define CELLF   50.0f
#define INV_CELLF (1.0f / 50.0f)

typedef int   v4i __attribute__((ext_vector_type(4)));
typedef float v4f __attribute__((ext_vector_type(4)));

// ---------------------------------------------------------------------------
// Zero the output accumulator (harness poisons d_out; we accumulate atomics).
// ---------------------------------------------------------------------------
__global__ __launch_bounds__(256) void zero_f32(float* __restrict__ out, int n) {
    int t = (blockIdx.x * blockDim.x + threadIdx.x) * 4;
    if (t + 3 < n) {
        *(v4f*)(out + t) = (v4f)0.0f;
    } else {
        for (int k = t; k < n; ++k) out[k] = 0.0f;
    }
}

// ---------------------------------------------------------------------------
// One pair: minimum-image, distance, LDS table lerp, scatter ±f via
// hardware float atomics (non-returning -> STOREcnt path at L2).
// ---------------------------------------------------------------------------
struct F3 { float x, y, z; };

__device__ __forceinline__ void do_pair(int i, int j,
                                        const float* __restrict__ q,
                                        float* __restrict__ force,
                                        const float* sF,
                                        float r0, float inv_dr)
{
    // 12-byte gathers; q (1.2 MB) stays L2-resident.
    F3 qi = *(const F3*)(q + 3 * (size_t)i);
    F3 qj = *(const F3*)(q + 3 * (size_t)j);

    float dx = qi.x - qj.x;
    float dy = qi.y - qj.y;
    float dz = qi.z - qj.z;

    // minimum image; jnp.round == round-half-even == v_rndne_f32
    dx -= CELLF * rintf(dx * INV_CELLF);
    dy -= CELLF * rintf(dy * INV_CELLF);
    dz -= CELLF * rintf(dz * INV_CELLF);

    float r2    = dx * dx + dy * dy + dz * dz;
    float r     = sqrtf(r2);
    float inv_r = 1.0f / fmaxf(r, 1e-12f);

    // Uniform-grid interp with jnp.interp endpoint clamping.
    float t = (r - r0) * inv_dr;
    t = fminf(fmaxf(t, 0.0f), (float)(N_TABLE - 1));
    int i0 = (int)t;
    i0 = (i0 > N_TABLE - 2) ? (N_TABLE - 2) : i0;
    float fr = t - (float)i0;
    float f0 = sF[i0];
    float f1 = sF[i0 + 1];
    float s  = (f0 + fr * (f1 - f0)) * inv_r;   // force magnitude / r

    float fx = s * dx, fy = s * dy, fz = s * dz;

    atomicAdd(force + 3 * (size_t)i + 0,  fx);
    atomicAdd(force + 3 * (size_t)i + 1,  fy);
    atomicAdd(force + 3 * (size_t)i + 2,  fz);
    atomicAdd(force + 3 * (size_t)j + 0, -fx);
    atomicAdd(force + 3 * (size_t)j + 1, -fy);
    atomicAdd(force + 3 * (size_t)j + 2, -fz);
}

// ---------------------------------------------------------------------------
// Main kernel: 2 pairs per thread via one NT b128 load of the pair stream.
// Table staged to LDS via CDNA5 async-to-LDS DMA (ASYNCcnt).
// ---------------------------------------------------------------------------
__global__ __launch_bounds__(256) void pair_forces(const float* __restrict__ q,
                                                   const int*   __restrict__ nbr,
                                                   const float* __restrict__ table_r,
                                                   const float* __restrict__ table_f,
                                                   float* __restrict__ force,
                                                   int n_pairs)
{
    __shared__ __align__(16) float sF[N_TABLE];

    // --- gfx1250 async global->LDS staging of the 4 KB table ---------------
    // 256 threads x b128 = 4096 B, one shot. VDST holds the LDS byte address
    // (low 32 bits of the flat pointer == LDS offset), VADDR the 64b global.
    {
        unsigned    lds_off = (unsigned)(unsigned long long)(&sF[0])
                              + (unsigned)threadIdx.x * 16u;
        const char* gsrc    = (const char*)table_f + (size_t)threadIdx.x * 16u;
        asm volatile("global_load_async_to_lds_b128 %0, %1, off"
                     :
                     : "v"(lds_off), "v"(gsrc)
                     : "memory");
        asm volatile("s_wait_asynccnt 0" ::: "memory");
    }
    __syncthreads();

    // Uniform table spacing (scalar loads; broadcast across the wave).
    float r0     = table_r[0];
    float inv_dr = (float)(N_TABLE - 1) / (table_r[N_TABLE - 1] - r0);

    int t  = blockIdx.x * blockDim.x + threadIdx.x;
    int p0 = t * 2;
    if (p0 + 1 < n_pairs) {
        // Non-temporal: 51 MB read-once stream must not evict q/force from L2.
        v4i pr = __builtin_nontemporal_load((const v4i*)nbr + t);
        do_pair(pr.x, pr.y, q, force, sF, r0, inv_dr);
        do_pair(pr.z, pr.w, q, force, sF, r0, inv_dr);
    } else if (p0 < n_pairs) {
        int i = nbr[2 * p0 + 0];
        int j = nbr[2 * p0 + 1];
        do_pair(i, j, q, force, sF, r0, inv_dr);
    }
}

// ---------------------------------------------------------------------------
extern "C" void kernel_launch(void* const* d_in, const int* in_sizes, int n_in,
                              void* d_out, int out_size, void* d_ws, size_t ws_size,
                              hipStream_t stream)
{
    const float* q       = (const float*)d_in[0];   // (N_ATOMS, 3) f32
    const int*   nbr     = (const int*)  d_in[1];   // (N_PAIRS, 2) i32
    const float* table_r = (const float*)d_in[2];   // (N_TABLE,)   f32
    const float* table_f = (const float*)d_in[3];   // (N_TABLE,)   f32
    float*       out     = (float*)d_out;           // (N_ATOMS, 3) f32

    int n_pairs = in_sizes[1] / 2;

    // 1) zero the accumulator
    int zthreads = (out_size + 3) / 4;
    zero_f32<<<(zthreads + 255) / 256, 256, 0, stream>>>(out, out_size);

    // 2) pair forces: 2 pairs per thread
    int nthreads = (n_pairs + 1) / 2;
    int nblocks  = (nthreads + 255) / 256;
    pair_forces<<<nblocks, 256, 0, stream>>>(q, nbr, table_r, table_f, out, n_pairs);
}